// GRAND_82884278878458
// MI455X (gfx1250) — compile-verified
//
#include <hip/hip_runtime.h>
#include <hip/hip_bf16.h>

typedef float v2f __attribute__((ext_vector_type(2)));
typedef float v8f __attribute__((ext_vector_type(8)));

#define IN_DIM   128
#define HID_DIM  256
#define N_CLASS  40
#define LOG_PITCH 48   // N_CLASS padded to 3x16 WMMA tiles

// ---------------------------------------------------------------- degree
__global__ void grand_deg_kernel(const int* __restrict__ dst, float* __restrict__ deg, int E) {
    int e = blockIdx.x * blockDim.x + threadIdx.x;
    if (e < E) unsafeAtomicAdd(&deg[dst[e]], 1.0f);
}

__global__ void grand_norm_kernel(float* __restrict__ deg, int n) {
    int i = blockIdx.x * blockDim.x + threadIdx.x;
    if (i < n) deg[i] = rsqrtf(fmaxf(deg[i], 1.0f));
}

// ---------------------------------------------------------------- SpMM: one wave32 per edge
// lane l moves float4 l of the 128-float row (32*4 = 128). Edge index is wave-uniform
// -> readfirstlane so src/dst/norm loads become scalar (SMEM) loads; x row gather is a
// coalesced global_load_b128; scatter via global_atomic_add_f32 (x fits in 192MB L2).
__global__ void grand_spmm_kernel(const float* __restrict__ x, float* __restrict__ xn,
                                  const int* __restrict__ src, const int* __restrict__ dst,
                                  const float* __restrict__ norm, int E) {
    int wid  = __builtin_amdgcn_readfirstlane((int)(threadIdx.x >> 5));
    int lane = threadIdx.x & 31;
    int e = blockIdx.x * 8 + wid;
    if (e >= E) return;
    int s = src[e];
    int d = dst[e];
    float w = norm[s] * norm[d];
    const float4 v = ((const float4*)(x + (size_t)s * IN_DIM))[lane];
    float* o = xn + (size_t)d * IN_DIM + lane * 4;
    unsafeAtomicAdd(o + 0, v.x * w);
    unsafeAtomicAdd(o + 1, v.y * w);
    unsafeAtomicAdd(o + 2, v.z * w);
    unsafeAtomicAdd(o + 3, v.w * w);
}

// ---------------------------------------------------------------- y += x (float4 grid-stride)
__global__ void grand_accum_kernel(float4* __restrict__ y, const float4* __restrict__ x, int n4) {
    int i = blockIdx.x * blockDim.x + threadIdx.x;
    int stride = gridDim.x * blockDim.x;
    for (; i < n4; i += stride) {
        float4 a = y[i], b = x[i];
        a.x += b.x; a.y += b.y; a.z += b.z; a.w += b.w;
        y[i] = a;
    }
}

// ---------------------------------------------------------------- GEMM1: h = relu((y/4) @ W1 + b1)
// Block: 256 thr = 8 waves, block tile 64(M) x 32(N), K=128 staged once in LDS.
// Each wave: one 16x16 tile via 32x V_WMMA_F32_16X16X4_F32.
// LDS pitches padded (132 / 33 floats) to spread rows across banks.
__global__ __launch_bounds__(256)
void grand_gemm1_kernel(const float* __restrict__ ysum, const float* __restrict__ W1,
                        const float* __restrict__ b1, float* __restrict__ h, int n_nodes) {
    __shared__ float As[64][132];
    __shared__ float Bs[128][33];

    const int rowBase = blockIdx.x * 64;
    const int colBase = blockIdx.y * 32;
    const int lane = threadIdx.x & 31;
    const int wid  = threadIdx.x >> 5;
    const int wm = wid & 3;        // 16-row tile in block
    const int wn = wid >> 2;       // 16-col tile in block
    const int half = lane >> 4;    // 0: K pair {0,1}, 1: K pair {2,3}
    const int lm = lane & 15;

    // stage A (64x128, pre-scaled by 1/(K+1)=0.25)
    #pragma unroll
    for (int it = 0; it < 8; ++it) {
        int li  = it * 256 + threadIdx.x;   // 0..2047 float4 slots
        int row = li >> 5;
        int c4  = (li & 31) * 4;
        float4 v = make_float4(0.f, 0.f, 0.f, 0.f);
        int grow = rowBase + row;
        if (grow < n_nodes) v = *(const float4*)(ysum + (size_t)grow * IN_DIM + c4);
        As[row][c4 + 0] = v.x * 0.25f;
        As[row][c4 + 1] = v.y * 0.25f;
        As[row][c4 + 2] = v.z * 0.25f;
        As[row][c4 + 3] = v.w * 0.25f;
    }
    // stage B (128x32 slice of W1)
    #pragma unroll
    for (int it = 0; it < 4; ++it) {
        int li  = it * 256 + threadIdx.x;   // 0..1023 float4 slots
        int row = li >> 3;
        int c4  = (li & 7) * 4;
        float4 v = *(const float4*)(W1 + (size_t)row * HID_DIM + colBase + c4);
        Bs[row][c4 + 0] = v.x;
        Bs[row][c4 + 1] = v.y;
        Bs[row][c4 + 2] = v.z;
        Bs[row][c4 + 3] = v.w;
    }
    __syncthreads();

    v8f acc = {};
    #pragma unroll
    for (int k0 = 0; k0 < IN_DIM; k0 += 4) {
        v2f a, b;
        a.x = As[16 * wm + lm][k0 + 2 * half + 0];
        a.y = As[16 * wm + lm][k0 + 2 * half + 1];
        b.x = Bs[k0 + 2 * half + 0][16 * wn + lm];
        b.y = Bs[k0 + 2 * half + 1][16 * wn + lm];
        acc = __builtin_amdgcn_wmma_f32_16x16x4_f32(false, a, false, b, (short)0, acc, false, false);
    }

    const int gcol = colBase + 16 * wn + lm;
    const float bias = b1[gcol];
    #pragma unroll
    for (int r = 0; r < 8; ++r) {
        int grow = rowBase + 16 * wm + r + 8 * half;
        if (grow < n_nodes) {
            float v = acc[r] + bias;
            h[(size_t)grow * HID_DIM + gcol] = v > 0.f ? v : 0.f;
        }
    }
}

// ---------------------------------------------------------------- GEMM2: logits = h @ W2 + b2
// Block: 128 thr = 4 waves, tile 64(M) x 48(N padded), K=256 in 4 chunks of 64.
// Each wave: 16 rows x 3 WMMA column tiles (cols 40..47 are exact zeros).
__global__ __launch_bounds__(128)
void grand_gemm2_kernel(const float* __restrict__ h, const float* __restrict__ W2,
                        const float* __restrict__ b2, float* __restrict__ logits, int n_nodes) {
    __shared__ float As[64][66];
    __shared__ float Bs[64][49];

    const int rowBase = blockIdx.x * 64;
    const int lane = threadIdx.x & 31;
    const int wid  = threadIdx.x >> 5;   // 0..3 -> 16-row tile
    const int half = lane >> 4;
    const int lm = lane & 15;

    v8f acc[3] = {};

    for (int kc = 0; kc < 4; ++kc) {
        const int kBase = kc * 64;
        // stage A chunk (64x64)
        #pragma unroll
        for (int it = 0; it < 8; ++it) {
            int li  = it * 128 + threadIdx.x;   // 0..1023 float4 slots
            int row = li >> 4;
            int c4  = (li & 15) * 4;
            float4 v = make_float4(0.f, 0.f, 0.f, 0.f);
            int grow = rowBase + row;
            if (grow < n_nodes) v = *(const float4*)(h + (size_t)grow * HID_DIM + kBase + c4);
            As[row][c4 + 0] = v.x;
            As[row][c4 + 1] = v.y;
            As[row][c4 + 2] = v.z;
            As[row][c4 + 3] = v.w;
        }
        // stage B chunk (64x48, zero-pad cols 40..47)
        #pragma unroll
        for (int it = 0; it < 24; ++it) {
            int li  = it * 128 + threadIdx.x;   // 0..3071
            int row = li / LOG_PITCH;
            int col = li % LOG_PITCH;
            Bs[row][col] = (col < N_CLASS) ? W2[(size_t)(kBase + row) * N_CLASS + col] : 0.f;
        }
        __syncthreads();

        #pragma unroll
        for (int k0 = 0; k0 < 64; k0 += 4) {
            v2f a;
            a.x = As[16 * wid + lm][k0 + 2 * half + 0];
            a.y = As[16 * wid + lm][k0 + 2 * half + 1];
            #pragma unroll
            for (int t = 0; t < 3; ++t) {
                v2f b;
                b.x = Bs[k0 + 2 * half + 0][16 * t + lm];
                b.y = Bs[k0 + 2 * half + 1][16 * t + lm];
                acc[t] = __builtin_amdgcn_wmma_f32_16x16x4_f32(false, a, false, b, (short)0, acc[t], false, false);
            }
        }
        __syncthreads();
    }

    #pragma unroll
    for (int t = 0; t < 3; ++t) {
        int col = 16 * t + lm;
        float bias = (col < N_CLASS) ? b2[col] : 0.f;
        #pragma unroll
        for (int r = 0; r < 8; ++r) {
            int grow = rowBase + 16 * wid + r + 8 * half;
            if (grow < n_nodes)
                logits[(size_t)grow * LOG_PITCH + col] = acc[t][r] + bias;
        }
    }
}

// ---------------------------------------------------------------- log_softmax over 40 classes
__global__ void grand_lsm_kernel(const float* __restrict__ logits, float* __restrict__ out, int n_nodes) {
    int r = blockIdx.x * blockDim.x + threadIdx.x;
    if (r >= n_nodes) return;
    const float* p = logits + (size_t)r * LOG_PITCH;
    float v[N_CLASS];
    float m = -INFINITY;
    #pragma unroll
    for (int c = 0; c < N_CLASS; ++c) { v[c] = p[c]; m = fmaxf(m, v[c]); }
    float s = 0.f;
    #pragma unroll
    for (int c = 0; c < N_CLASS; ++c) s += expf(v[c] - m);
    float lse = m + logf(s);
    #pragma unroll
    for (int c = 0; c < N_CLASS; ++c) out[(size_t)r * N_CLASS + c] = v[c] - lse;
}

// ---------------------------------------------------------------- launch
extern "C" void kernel_launch(void* const* d_in, const int* in_sizes, int n_in,
                              void* d_out, int out_size, void* d_ws, size_t ws_size,
                              hipStream_t stream) {
    const float* feats = (const float*)d_in[0];
    const int*   src   = (const int*)d_in[1];
    const int*   dst   = (const int*)d_in[2];
    const float* W1    = (const float*)d_in[3];
    const float* b1    = (const float*)d_in[4];
    const float* W2    = (const float*)d_in[5];
    const float* b2    = (const float*)d_in[6];
    float* out = (float*)d_out;

    const int N = in_sizes[0] / IN_DIM;
    const int E = in_sizes[1];

    // workspace carve-up (256B aligned)
    char* ws = (char*)d_ws;
    size_t off = 0;
    auto take = [&](size_t bytes) -> char* {
        char* p = ws + off;
        off = (off + bytes + 255) & ~(size_t)255;
        return p;
    };
    const size_t xbytes = (size_t)N * IN_DIM * sizeof(float);
    float* norm   = (float*)take((size_t)N * sizeof(float));
    float* xA     = (float*)take(xbytes);
    float* xB     = (float*)take(xbytes);
    float* ysum   = (float*)take(xbytes);
    float* hbuf   = (float*)take((size_t)N * HID_DIM * sizeof(float));
    float* logits = (float*)take((size_t)N * LOG_PITCH * sizeof(float));
    (void)ws_size; (void)n_in; (void)out_size;

    // degrees -> norm = clip(deg,1)^-0.5
    hipMemsetAsync(norm, 0, (size_t)N * sizeof(float), stream);
    grand_deg_kernel<<<(E + 255) / 256, 256, 0, stream>>>(dst, norm, E);
    grand_norm_kernel<<<(N + 255) / 256, 256, 0, stream>>>(norm, N);

    // x = feats; y = feats
    hipMemcpyAsync(xA,   feats, xbytes, hipMemcpyDeviceToDevice, stream);
    hipMemcpyAsync(ysum, feats, xbytes, hipMemcpyDeviceToDevice, stream);

    // K=3 propagation steps: xnew = scatter(x[src]*w, dst); y += xnew
    const int n4 = N * IN_DIM / 4;
    float* xin = xA;
    float* xout = xB;
    for (int step = 0; step < 3; ++step) {
        hipMemsetAsync(xout, 0, xbytes, stream);
        grand_spmm_kernel<<<(E + 7) / 8, 256, 0, stream>>>(xin, xout, src, dst, norm, E);
        grand_accum_kernel<<<2048, 256, 0, stream>>>((float4*)ysum, (const float4*)xout, n4);
        float* t = xin; xin = xout; xout = t;
    }

    // MLP head on WMMA
    dim3 g1((N + 63) / 64, HID_DIM / 32);
    grand_gemm1_kernel<<<g1, 256, 0, stream>>>(ysum, W1, b1, hbuf, N);
    grand_gemm2_kernel<<<(N + 63) / 64, 128, 0, stream>>>(hbuf, W2, b2, logits, N);
    grand_lsm_kernel<<<(N + 255) / 256, 256, 0, stream>>>(logits, out, N);
}